// SMART_5385888989441
// MI455X (gfx1250) — compile-verified
//
#include <hip/hip_runtime.h>
#include <hip/hip_bf16.h>

typedef __attribute__((ext_vector_type(2))) float v2f;
typedef __attribute__((ext_vector_type(8))) float v8f;

#define EPSF 1e-8f

// ---------------------------------------------------------------------------
// Per-node L2 norms: one wave per node, float4 loads, shfl reduction.
// ---------------------------------------------------------------------------
template <int D>
__global__ void node_norms(const float* __restrict__ x, float* __restrict__ nrm, int N) {
    constexpr int V = D / 128;  // float4 chunks per lane
    int n = (int)((blockIdx.x * blockDim.x + threadIdx.x) >> 5);
    int lane = threadIdx.x & 31;
    if (n >= N) return;
    const float4* row = (const float4*)(x + (size_t)n * D);
    float ss = 0.f;
#pragma unroll
    for (int v = 0; v < V; ++v) {
        float4 a = row[lane + 32 * v];
        ss += a.x * a.x + a.y * a.y + a.z * a.z + a.w * a.w;
    }
#pragma unroll
    for (int off = 16; off > 0; off >>= 1) ss += __shfl_xor(ss, off, 32);
    if (lane == 0) nrm[n] = sqrtf(ss);
}

// ---------------------------------------------------------------------------
// agg initialization = node features (self-loops always pass: cos(x,x)=1>0.5;
// zero rows contribute zero either way). Grid-stride float4 copy.
// ---------------------------------------------------------------------------
__global__ void copy_f4(const float4* __restrict__ in, float4* __restrict__ out, size_t n4) {
    size_t i = (size_t)blockIdx.x * blockDim.x + threadIdx.x;
    size_t stride = (size_t)gridDim.x * blockDim.x;
    for (; i < n4; i += stride) out[i] = in[i];
}

// ---------------------------------------------------------------------------
// Edge kernel: one wave per edge. Dot product of endpoint rows via float4
// loads + shfl-xor reduce; wave-uniform similarity gate; masked scatter-add
// of the SOURCE row into agg[dst] via global_atomic_add_f32 (L2-resident).
// ---------------------------------------------------------------------------
template <int D>
__global__ void edge_conv(const float* __restrict__ x, const float* __restrict__ nrm,
                          const int* __restrict__ src, const int* __restrict__ dst,
                          float* __restrict__ agg, int E) {
    constexpr int V = D / 128;
    int e = (int)((blockIdx.x * blockDim.x + threadIdx.x) >> 5);
    int lane = threadIdx.x & 31;
    if (e >= E) return;
    int s = src[e];
    int d = dst[e];
    const float4* xs = (const float4*)(x + (size_t)s * D);  // x_j (source / message)
    const float4* xd = (const float4*)(x + (size_t)d * D);  // x_i (target)
    float4 aj[V];
    float dot = 0.f;
#pragma unroll
    for (int v = 0; v < V; ++v) {
        float4 a = xs[lane + 32 * v];
        float4 b = xd[lane + 32 * v];
        aj[v] = a;
        dot += a.x * b.x + a.y * b.y + a.z * b.z + a.w * b.w;
    }
#pragma unroll
    for (int off = 16; off > 0; off >>= 1) dot += __shfl_xor(dot, off, 32);
    float denom = fmaxf(nrm[s] * nrm[d], EPSF);
    if (dot > 0.5f * denom) {  // wave-uniform branch: EXEC stays full
        float* ad = agg + (size_t)d * D;
#pragma unroll
        for (int v = 0; v < V; ++v) {
            int base = (lane + 32 * v) * 4;
            atomicAdd(ad + base + 0, aj[v].x);
            atomicAdd(ad + base + 1, aj[v].y);
            atomicAdd(ad + base + 2, aj[v].z);
            atomicAdd(ad + base + 3, aj[v].w);
        }
    }
}

// ---------------------------------------------------------------------------
// FP32 WMMA GEMM: C[M x NOUT] = A[M x K] @ W^T + bias, optional ReLU.
// W stored row-major [NOUT x K] (PyTorch-style), so B[k][n] = W[n][k].
// One wave computes a 16x16 C tile; K-loop in steps of 4 with
// V_WMMA_F32_16X16X4_F32. M must be a multiple of 16 (50000 = 3125*16).
//
// A-frag (16x4 f32, 2 VGPRs): lanes 0-15 -> rows, K=0/1; lanes 16-31 -> K=2/3.
// B-frag mirrored. C/D (16x16 f32, 8 VGPRs): VGPR r -> rows r (lanes 0-15)
// and r+8 (lanes 16-31), N = lane&15.
// ---------------------------------------------------------------------------
template <int K, int NOUT, bool RELU>
__global__ void gemm_wmma_f32(const float* __restrict__ A, const float* __restrict__ W,
                              const float* __restrict__ bias, float* __restrict__ C, int M) {
    int wave = (int)((blockIdx.x * blockDim.x + threadIdx.x) >> 5);
    int lane = threadIdx.x & 31;
    constexpr int NT = NOUT / 16;
    int mt = wave / NT;
    int nt = wave % NT;
    if (mt * 16 >= M) return;  // wave-uniform: EXEC all-1s inside
    int half = lane >> 4;      // 0: K pair {0,1}; 1: K pair {2,3}
    int l = lane & 15;

    const float* arow = A + (size_t)(mt * 16 + l) * K;
    const float* wrow = W + (size_t)(nt * 16 + l) * K;

    v8f acc = {};
    for (int k0 = 0; k0 < K; k0 += 4) {
        v2f a, b;
        a.x = arow[k0 + 2 * half];
        a.y = arow[k0 + 2 * half + 1];
        b.x = wrow[k0 + 2 * half];
        b.y = wrow[k0 + 2 * half + 1];
        acc = __builtin_amdgcn_wmma_f32_16x16x4_f32(
            /*neg_a=*/false, a, /*neg_b=*/false, b,
            /*c_mod=*/(short)0, acc, /*reuse_a=*/false, /*reuse_b=*/false);
    }

    float bv = bias[nt * 16 + l];
#pragma unroll
    for (int r = 0; r < 8; ++r) {
        float vv = acc[r] + bv;
        if (RELU) vv = fmaxf(vv, 0.f);
        C[(size_t)(mt * 16 + half * 8 + r) * NOUT + nt * 16 + l] = vv;
    }
}

// ---------------------------------------------------------------------------
// log_softmax over 64 logits: one wave per node, 2 values per lane.
// ---------------------------------------------------------------------------
__global__ void log_softmax64(const float* __restrict__ z, float* __restrict__ out, int N) {
    int n = (int)((blockIdx.x * blockDim.x + threadIdx.x) >> 5);
    int lane = threadIdx.x & 31;
    if (n >= N) return;
    const float* row = z + (size_t)n * 64;
    float v0 = row[lane];
    float v1 = row[lane + 32];
    float m = fmaxf(v0, v1);
#pragma unroll
    for (int off = 16; off > 0; off >>= 1) m = fmaxf(m, __shfl_xor(m, off, 32));
    float s = __expf(v0 - m) + __expf(v1 - m);
#pragma unroll
    for (int off = 16; off > 0; off >>= 1) s += __shfl_xor(s, off, 32);
    float lse = m + __logf(s);
    float* orow = out + (size_t)n * 64;
    orow[lane] = v0 - lse;
    orow[lane + 32] = v1 - lse;
}

// ---------------------------------------------------------------------------
// Host launcher.
// ---------------------------------------------------------------------------
extern "C" void kernel_launch(void* const* d_in, const int* in_sizes, int n_in,
                              void* d_out, int out_size, void* d_ws, size_t ws_size,
                              hipStream_t stream) {
    constexpr int D_IN = 128, D_HID = 256, D_OUT = 64;
    const float* x  = (const float*)d_in[0];
    const int*   ei = (const int*)d_in[1];  // edge_index, [2 x E] flat
    const float* W1 = (const float*)d_in[2];
    const float* b1 = (const float*)d_in[3];
    const float* W2 = (const float*)d_in[4];
    const float* b2 = (const float*)d_in[5];
    float* out = (float*)d_out;

    const int N = in_sizes[0] / D_IN;
    const int E = in_sizes[1] / 2;
    const int* src = ei;
    const int* dst = ei + E;

    // Workspace carve-out (256B aligned slabs).
    auto align256 = [](size_t v) { return (v + 255) & ~(size_t)255; };
    char* ws = (char*)d_ws;
    size_t off = 0;
    float* nrm1 = (float*)(ws + off); off += align256((size_t)N * sizeof(float));
    float* agg1 = (float*)(ws + off); off += align256((size_t)N * D_IN * sizeof(float));
    float* h    = (float*)(ws + off); off += align256((size_t)N * D_HID * sizeof(float));
    float* nrm2 = (float*)(ws + off); off += align256((size_t)N * sizeof(float));
    float* agg2 = (float*)(ws + off); off += align256((size_t)N * D_HID * sizeof(float));
    float* z    = (float*)(ws + off); off += align256((size_t)N * D_OUT * sizeof(float));
    (void)ws_size; (void)n_in; (void)out_size;

    const int THREADS = 256;                // 8 waves per block
    const int wavesPerBlk = THREADS / 32;
    int nodeBlks = (N + wavesPerBlk - 1) / wavesPerBlk;
    int edgeBlks = (E + wavesPerBlk - 1) / wavesPerBlk;

    // ---- Layer 1 ----
    node_norms<D_IN><<<nodeBlks, THREADS, 0, stream>>>(x, nrm1, N);
    {
        size_t n4 = (size_t)N * D_IN / 4;
        int blks = (int)((n4 + THREADS - 1) / THREADS);
        if (blks > 65535) blks = 65535;
        copy_f4<<<blks, THREADS, 0, stream>>>((const float4*)x, (float4*)agg1, n4);
    }
    edge_conv<D_IN><<<edgeBlks, THREADS, 0, stream>>>(x, nrm1, src, dst, agg1, E);
    {
        int waves = (N / 16) * (D_HID / 16);
        int blks = (waves * 32 + 127) / 128;
        gemm_wmma_f32<D_IN, D_HID, true><<<blks, 128, 0, stream>>>(agg1, W1, b1, h, N);
    }

    // ---- Layer 2 ----
    node_norms<D_HID><<<nodeBlks, THREADS, 0, stream>>>(h, nrm2, N);
    {
        size_t n4 = (size_t)N * D_HID / 4;
        int blks = (int)((n4 + THREADS - 1) / THREADS);
        if (blks > 65535) blks = 65535;
        copy_f4<<<blks, THREADS, 0, stream>>>((const float4*)h, (float4*)agg2, n4);
    }
    edge_conv<D_HID><<<edgeBlks, THREADS, 0, stream>>>(h, nrm2, src, dst, agg2, E);
    {
        int waves = (N / 16) * (D_OUT / 16);
        int blks = (waves * 32 + 127) / 128;
        gemm_wmma_f32<D_HID, D_OUT, false><<<blks, 128, 0, stream>>>(agg2, W2, b2, z, N);
    }

    // ---- log_softmax ----
    log_softmax64<<<nodeBlks, THREADS, 0, stream>>>(z, out, N);
}